// GuidedResampler_455266534007
// MI455X (gfx1250) — compile-verified
//
#include <hip/hip_runtime.h>
#include <math.h>

// GuidedResampler for MI455X (gfx1250, wave32).
// Output is constant within each 4x4 high-res cell, so we compute per-cell
// values via top-2 + softmax + weighted block sums, expressed as a
// 16x128 = (16x32)x(32x128) matmul using V_WMMA_F32_16X16X4_F32 chains.

#define B_    4
#define C_    128
#define H_    128
#define W_    128
#define HL_   32
#define WL_   32
#define NLOW  1024     // HL_*WL_
#define NHIGH 16384

typedef __attribute__((ext_vector_type(2))) float v2f;
typedef __attribute__((ext_vector_type(8))) float v8f;

// ---------------------------------------------------------------------------
// Kernel 1: S[b, block, c] = sum of the 4x4 patch of v[b, c, :, :] at block.
// One thread per (b, c, mr, mc), mc fastest -> float4 reads fully coalesced.
// ---------------------------------------------------------------------------
__global__ void gr_block_sum(const float* __restrict__ v, float* __restrict__ S) {
    int p  = blockIdx.x * blockDim.x + threadIdx.x;
    int mc = p & 31;
    int mr = (p >> 5) & 31;
    int c  = (p >> 10) & (C_ - 1);
    int b  = p >> 17;
    const float* base = v + (((size_t)(b * C_ + c) * H_) + mr * 4) * W_ + mc * 4;
    float s = 0.f;
#pragma unroll
    for (int r = 0; r < 4; ++r) {
        float4 q = *(const float4*)(base + r * W_);
        s += q.x + q.y + q.z + q.w;
    }
    S[((size_t)(b * NLOW + mr * WL_ + mc)) * C_ + c] = s;
}

// ---------------------------------------------------------------------------
// Kernel 2: per row (b,n) of the 1024-wide coarse attention map:
// top-2 (lower index wins ties, matching jax.lax.top_k) + 2-way softmax.
// One 256-thread block per row; LDS tree reduction.
// ---------------------------------------------------------------------------
__device__ __forceinline__ void top2_insert(float v, int j,
                                            float& v1, int& i1,
                                            float& v2, int& i2) {
    if (v > v1 || (v == v1 && j < i1)) { v2 = v1; i2 = i1; v1 = v; i1 = j; }
    else if (v > v2 || (v == v2 && j < i2)) { v2 = v; i2 = j; }
}

__global__ void gr_top2(const float* __restrict__ attn,
                        float* __restrict__ wbuf, int* __restrict__ ibuf) {
    __shared__ float rv1[256], rv2[256];
    __shared__ int   ri1[256], ri2[256];
    const int row = blockIdx.x;          // b*NLOW + n
    const int tid = threadIdx.x;
    const float* rp = attn + (size_t)row * NLOW;

    float v1 = -INFINITY, v2 = -INFINITY;
    int   i1 = 0x7fffffff, i2 = 0x7fffffff;
#pragma unroll
    for (int s = 0; s < 4; ++s) {
        int j = tid + s * 256;
        top2_insert(rp[j], j, v1, i1, v2, i2);
    }
    rv1[tid] = v1; ri1[tid] = i1; rv2[tid] = v2; ri2[tid] = i2;
    __syncthreads();
    for (int off = 128; off > 0; off >>= 1) {
        if (tid < off) {
            top2_insert(rv1[tid + off], ri1[tid + off], v1, i1, v2, i2);
            top2_insert(rv2[tid + off], ri2[tid + off], v1, i1, v2, i2);
            rv1[tid] = v1; ri1[tid] = i1; rv2[tid] = v2; ri2[tid] = i2;
        }
        __syncthreads();
    }
    if (tid == 0) {
        float e   = __expf(v2 - v1);     // <= 1, stable
        float inv = 1.0f / (1.0f + e);
        wbuf[row * 2 + 0] = inv;
        wbuf[row * 2 + 1] = e * inv;
        ibuf[row * 2 + 0] = i1;
        ibuf[row * 2 + 1] = i2;
    }
}

// ---------------------------------------------------------------------------
// Kernel 3: per (b, group of 16 cells): D(16x128) = A(16x32) * B(32x128)
// A[m, 2m+kk] = w[b, n0+m, kk]/16 (sparse), B rows = gathered S rows.
// 256 threads = 8 waves; wave w computes the 16-channel tile w via 8 chained
// V_WMMA_F32_16X16X4_F32. Per-lane, A's two nonzeros (cols 2m, 2m+1) live in
// exactly one K-chunk kc* = m>>1 and only on the lane-half with matching
// parity -> hoist the sw loads, select with v_cndmask inside the loop
// (no divergent LDS loads, no exec save/restore in the hot loop).
// D staged in LDS, then coalesced float4-splat writes.
// ---------------------------------------------------------------------------
__global__ void gr_combine(const float* __restrict__ S,
                           const float* __restrict__ wbuf,
                           const int* __restrict__ ibuf,
                           float* __restrict__ out) {
    __shared__ float sw[32];
    __shared__ int   sidx[32];
    __shared__ __align__(16) float Bl[32 * C_];  // 16 KB gathered S rows
    __shared__ __align__(16) float Ol[16 * C_];  //  8 KB out_low (cell x ch)

    const int tid = threadIdx.x;
    const int g   = blockIdx.x;     // b*64 + ct
    const int b   = g >> 6;
    const int ct  = g & 63;
    const int n0  = ct * 16;

    if (tid < 32) {
        int cell = tid >> 1, kk = tid & 1;
        int base = (b * NLOW + n0 + cell) * 2 + kk;
        sidx[tid] = ibuf[base];
        sw[tid]   = wbuf[base] * (1.0f / 16.0f);
    }
    __syncthreads();

    // Gather 32 rows x 128 ch of S into LDS, float4-vectorized (b128 both ways)
    for (int p = tid; p < 32 * 32; p += 256) {       // 1024 float4s
        int rowj = p >> 5, q4 = p & 31;
        const float4* src =
            (const float4*)(S + ((size_t)(b * NLOW + sidx[rowj])) * C_);
        ((float4*)Bl)[p] = src[q4];
    }
    __syncthreads();

    const int wave  = tid >> 5;
    const int lane  = tid & 31;
    const int m     = lane & 15;     // A row (M); also N index for B/D
    const int half  = lane >> 4;
    const int koff  = half * 2;
    const int ch    = wave * 16 + m; // channel column (N) for B and D

    // Hoisted sparse-A fragment: nonzero only when half == parity(m), at kc*.
    const bool  act   = (half == (m & 1));
    const int   kstar = m >> 1;
    const float ax    = act ? sw[2 * m]     : 0.0f;
    const float ay    = act ? sw[2 * m + 1] : 0.0f;

    v8f acc = {};
#pragma unroll
    for (int kc = 0; kc < 8; ++kc) {
        const int c0 = 4 * kc + koff;
        v2f a, bb;
        a.x  = (kc == kstar) ? ax : 0.0f;    // v_cndmask, no LDS
        a.y  = (kc == kstar) ? ay : 0.0f;
        bb.x = Bl[c0 * C_ + ch];             // B rows c0, c0+1, column ch
        bb.y = Bl[(c0 + 1) * C_ + ch];
        acc = __builtin_amdgcn_wmma_f32_16x16x4_f32(
            false, a, false, bb, (short)0, acc, false, false);
    }

    // D layout: VGPR r, lanes<16 -> cell r, lanes>=16 -> cell r+8
#pragma unroll
    for (int r = 0; r < 8; ++r) {
        Ol[(r + half * 8) * C_ + ch] = acc[r];
    }
    __syncthreads();

    // Scatter: group covers rows mr*4..+3, cols mc0*4..mc0*4+63, all channels.
    const int mr  = n0 >> 5;
    const int mc0 = n0 & 31;
    for (int p = tid; p < 8192; p += 256) {
        int cq = p & 15;            // which cell (column quad)
        int r  = (p >> 4) & 3;      // row within cell
        int c  = p >> 6;            // channel
        float val = Ol[cq * C_ + c];
        size_t idx = (((size_t)(b * C_ + c) * H_) + mr * 4 + r) * W_
                     + mc0 * 4 + cq * 4;
        float4 q; q.x = val; q.y = val; q.z = val; q.w = val;
        *(float4*)(out + idx) = q;
    }
}

// ---------------------------------------------------------------------------
extern "C" void kernel_launch(void* const* d_in, const int* in_sizes, int n_in,
                              void* d_out, int out_size, void* d_ws, size_t ws_size,
                              hipStream_t stream) {
    const float* v    = (const float*)d_in[0];  // (B,C,H,W) f32
    const float* attn = (const float*)d_in[1];  // (B,NLOW,NLOW) f32
    float* out = (float*)d_out;                 // (B,C,H,W) f32

    char* ws = (char*)d_ws;
    float* S    = (float*)ws;                                   // 2 MB
    size_t offS = (size_t)B_ * NLOW * C_ * sizeof(float);
    float* wbuf = (float*)(ws + offS);                          // 32 KB
    size_t offW = offS + (size_t)B_ * NLOW * 2 * sizeof(float);
    int*   ibuf = (int*)(ws + offW);                            // 32 KB

    gr_block_sum<<<(B_ * C_ * NLOW) / 256, 256, 0, stream>>>(v, S);
    gr_top2<<<B_ * NLOW, 256, 0, stream>>>(attn, wbuf, ibuf);
    gr_combine<<<B_ * (NLOW / 16), 256, 0, stream>>>(S, wbuf, ibuf, out);
}